// SAGE_3212635537937
// MI455X (gfx1250) — compile-verified
//
#include <hip/hip_runtime.h>
#include <hip/hip_bf16.h>

typedef __attribute__((ext_vector_type(2))) float v2f;
typedef __attribute__((ext_vector_type(8))) float v8f;

#define DDIM 128

// ---------------------------------------------------------------------------
// deg[dst] += 1 over all edges (layer-invariant)
// ---------------------------------------------------------------------------
__global__ void sage_deg_kernel(const long long* __restrict__ dst,
                                float* __restrict__ deg, int E) {
    int e = blockIdx.x * blockDim.x + threadIdx.x;
    if (e < E) atomicAdd(&deg[(int)dst[e]], 1.0f);
}

// deg -> invdeg = 1 / max(deg, 1)
__global__ void sage_invdeg_kernel(float* __restrict__ deg, int n) {
    int i = blockIdx.x * blockDim.x + threadIdx.x;
    if (i < n) deg[i] = 1.0f / fmaxf(deg[i], 1.0f);
}

// ---------------------------------------------------------------------------
// agg[dst,:] += x[src,:]  — one wave (32 lanes) per edge, float4 per lane
// ---------------------------------------------------------------------------
__global__ void sage_scatter_kernel(const float* __restrict__ x,
                                    const long long* __restrict__ src,
                                    const long long* __restrict__ dst,
                                    float* __restrict__ agg, int E) {
    int edge = (blockIdx.x * blockDim.x + threadIdx.x) >> 5;  // wave id
    int lane = threadIdx.x & 31;
    if (edge >= E) return;  // uniform per wave
    int s = (int)src[edge];
    int t = (int)dst[edge];
    const float4 v = *(const float4*)(x + (size_t)s * DDIM + lane * 4);
    float* o = agg + (size_t)t * DDIM + lane * 4;
    atomicAdd(o + 0, v.x);
    atomicAdd(o + 1, v.y);
    atomicAdd(o + 2, v.z);
    atomicAdd(o + 3, v.w);
}

// ---------------------------------------------------------------------------
// out = relu( (invdeg .* agg) @ Wl^T + b + xin @ Wr^T )
// One wave per 16x16 tile, f32 WMMA 16x16x4, K-loop of 32.
// A 16x4 layout: lane -> M = lane%16, VGPRs hold K pair, lanes>=16 at K+2.
// B 4x16 layout: lane -> N = lane%16, same K striping.  B[k][n] = W[n][k].
// C/D: VGPR v -> row v + 8*(lane>=16), col lane%16.
// ---------------------------------------------------------------------------
__global__ void __launch_bounds__(256)
sage_gemm_kernel(const float* __restrict__ xin, const float* __restrict__ agg,
                 const float* __restrict__ invdeg, const float* __restrict__ Wl,
                 const float* __restrict__ bias, const float* __restrict__ Wr,
                 float* __restrict__ out) {
    const int lane = threadIdx.x & 31;
    const int wave = threadIdx.x >> 5;
    const int row0 = blockIdx.x * 16;       // 16 output rows per block
    const int col0 = wave * 16;             // 8 waves cover 128 output cols
    const int mn   = lane & 15;             // M for A, N for B
    const int kh   = (lane >> 4) << 1;      // 0 or 2 : K sub-offset per half-wave

    const float* aggRow = agg + (size_t)(row0 + mn) * DDIM + kh;
    const float* xRow   = xin + (size_t)(row0 + mn) * DDIM + kh;
    const float* wlRow  = Wl  + (size_t)(col0 + mn) * DDIM + kh;  // W[n][k] row-major
    const float* wrRow  = Wr  + (size_t)(col0 + mn) * DDIM + kh;

    v8f accl = {};  // agg @ Wl^T  (unscaled; scale factored to epilogue)
    v8f accr = {};  // x   @ Wr^T

#pragma unroll
    for (int kt = 0; kt < DDIM / 4; ++kt) {
        v2f aA = *(const v2f*)(aggRow + kt * 4);
        v2f bL = *(const v2f*)(wlRow + kt * 4);
        accl = __builtin_amdgcn_wmma_f32_16x16x4_f32(false, aA, false, bL,
                                                     (short)0, accl, false, false);
        v2f aX = *(const v2f*)(xRow + kt * 4);
        v2f bR = *(const v2f*)(wrRow + kt * 4);
        accr = __builtin_amdgcn_wmma_f32_16x16x4_f32(false, aX, false, bR,
                                                     (short)0, accr, false, false);
    }

    const float bv   = bias[col0 + mn];
    const int   roff = (lane >> 4) << 3;  // 0 or 8
#pragma unroll
    for (int v = 0; v < 8; ++v) {
        const int r = row0 + roff + v;
        float val = invdeg[r] * accl[v] + bv + accr[v];
        out[(size_t)r * DDIM + col0 + mn] = fmaxf(val, 0.0f);
    }
}

// ---------------------------------------------------------------------------
extern "C" void kernel_launch(void* const* d_in, const int* in_sizes, int n_in,
                              void* d_out, int out_size, void* d_ws, size_t ws_size,
                              hipStream_t stream) {
    const float*     x   = (const float*)d_in[0];
    const long long* ei  = (const long long*)d_in[1];   // int64 (2, E)
    // d_in[2] = edge_weight : unused by the reference computation
    const float* W1l = (const float*)d_in[3];
    const float* b1  = (const float*)d_in[4];
    const float* W1r = (const float*)d_in[5];
    const float* W2l = (const float*)d_in[6];
    const float* b2  = (const float*)d_in[7];
    const float* W2r = (const float*)d_in[8];
    const float* W3l = (const float*)d_in[9];
    const float* b3  = (const float*)d_in[10];
    const float* W3r = (const float*)d_in[11];

    const int N = in_sizes[0] / DDIM;      // 100000
    const int E = in_sizes[1] / 2;         // 1600000
    const long long* src = ei;
    const long long* dst = ei + E;

    // Workspace layout
    float* agg = (float*)d_ws;                         // N*128 f32
    float* hB  = agg + (size_t)N * DDIM;               // N*128 f32
    float* deg = hB + (size_t)N * DDIM;                // N     f32  (-> invdeg)
    float* out = (float*)d_out;

    const size_t featBytes = (size_t)N * DDIM * sizeof(float);

    // Degree (layer-invariant) -> invdeg
    hipMemsetAsync(deg, 0, (size_t)N * sizeof(float), stream);
    sage_deg_kernel<<<(E + 255) / 256, 256, 0, stream>>>(dst, deg, E);
    sage_invdeg_kernel<<<(N + 255) / 256, 256, 0, stream>>>(deg, N);

    const int scatterBlocks = (E * 32 + 255) / 256;    // one wave per edge
    const int gemmBlocks    = N / 16;                  // N divisible by 16

    // Layer 1: in = x, out = d_out
    hipMemsetAsync(agg, 0, featBytes, stream);
    sage_scatter_kernel<<<scatterBlocks, 256, 0, stream>>>(x, src, dst, agg, E);
    sage_gemm_kernel<<<gemmBlocks, 256, 0, stream>>>(x, agg, deg, W1l, b1, W1r, out);

    // Layer 2: in = d_out, out = hB
    hipMemsetAsync(agg, 0, featBytes, stream);
    sage_scatter_kernel<<<scatterBlocks, 256, 0, stream>>>(out, src, dst, agg, E);
    sage_gemm_kernel<<<gemmBlocks, 256, 0, stream>>>(out, agg, deg, W2l, b2, W2r, hB);

    // Layer 3: in = hB, out = d_out
    hipMemsetAsync(agg, 0, featBytes, stream);
    sage_scatter_kernel<<<scatterBlocks, 256, 0, stream>>>(hB, src, dst, agg, E);
    sage_gemm_kernel<<<gemmBlocks, 256, 0, stream>>>(hB, agg, deg, W3l, b3, W3r, out);
}